// SubdivideMeshes_68839735821120
// MI455X (gfx1250) — compile-verified
//
#include <hip/hip_runtime.h>

typedef float v2f __attribute__((ext_vector_type(2)));
typedef float v8f __attribute__((ext_vector_type(8)));

#define NEG_SLOPE 0.01f

__device__ __forceinline__ void atomAddF(float* p, float v) {
  __hip_atomic_fetch_add(p, v, __ATOMIC_RELAXED, __HIP_MEMORY_SCOPE_AGENT);
}

// ---- degree / normalization (graph identical across batches) ----
__global__ void count_deg_k(const int* __restrict__ edges, int* __restrict__ deg, int E) {
  int t = blockIdx.x * blockDim.x + threadIdx.x;
  if (t < E)
    __hip_atomic_fetch_add(&deg[edges[2 * t + 1]], 1, __ATOMIC_RELAXED, __HIP_MEMORY_SCOPE_AGENT);
}

__global__ void dinv_k(const int* __restrict__ deg, float* __restrict__ dinv, int Vn) {
  int t = blockIdx.x * blockDim.x + threadIdx.x;
  if (t < Vn) dinv[t] = rsqrtf((float)(deg[t] + 1));  // +1 self loop; always > 0
}

// ---- conv1 GEMM: T[N x 16] = X[N x 3] @ W1[3 x 16], K padded 3->4, 1 WMMA/tile ----
__global__ void conv1_wmma_k(const float* __restrict__ x, const float* __restrict__ W1,
                             float* __restrict__ T, int N) {
  int lane = threadIdx.x & 31;
  int wave = threadIdx.x >> 5;
  int tile = blockIdx.x * (blockDim.x >> 5) + wave;
  int base = tile * 16;
  if (base >= N) return;                 // wave-uniform: EXEC stays all-ones for WMMA
  int half = lane >> 4;                  // 0: K{0,1}, 1: K{2,3}
  int l = lane & 15;
  int row = base + l;
  int k0 = 2 * half, k1 = 2 * half + 1;
  v2f a, b;
  a.x = x[(size_t)row * 3 + k0];                         // k0 in {0,2}, always < 3
  a.y = (k1 < 3) ? x[(size_t)row * 3 + k1] : 0.0f;       // pad K=3 with zeros
  b.x = W1[k0 * 16 + l];
  b.y = (k1 < 3) ? W1[k1 * 16 + l] : 0.0f;
  v8f c = {};
  c = __builtin_amdgcn_wmma_f32_16x16x4_f32(false, a, false, b, (short)0, c, false, false);
#pragma unroll
  for (int r = 0; r < 8; ++r) {
    int n = base + r + 8 * half;         // D layout: VGPR r -> row r (lanes 0-15) / r+8 (16-31)
    T[(size_t)n * 16 + l] = c[r];
  }
}

// ---- conv2 GEMM: T[N x 16] = H[N x 16] @ W2[16 x 16], 4 chained WMMAs/tile ----
__global__ void conv2_wmma_k(const float* __restrict__ h, const float* __restrict__ W2,
                             float* __restrict__ T, int N) {
  int lane = threadIdx.x & 31;
  int wave = threadIdx.x >> 5;
  int tile = blockIdx.x * (blockDim.x >> 5) + wave;
  int base = tile * 16;
  if (base >= N) return;
  int half = lane >> 4;
  int l = lane & 15;
  int row = base + l;
  v8f c = {};
#pragma unroll
  for (int kc = 0; kc < 4; ++kc) {
    int k0 = 4 * kc + 2 * half;
    v2f a, b;
    a.x = h[(size_t)row * 16 + k0];
    a.y = h[(size_t)row * 16 + k0 + 1];
    b.x = W2[k0 * 16 + l];
    b.y = W2[(k0 + 1) * 16 + l];
    c = __builtin_amdgcn_wmma_f32_16x16x4_f32(false, a, false, b, (short)0, c, false, false);
  }
#pragma unroll
  for (int r = 0; r < 8; ++r) {
    int n = base + r + 8 * half;
    T[(size_t)n * 16 + l] = c[r];
  }
}

// ---- conv3 GEMV: T3[N x 3] = H[N x 16] @ W3[16 x 3] (scalar: WMMA would waste 13/16 lanes) ----
__global__ void conv3_k(const float* __restrict__ h, const float* __restrict__ W3,
                        float* __restrict__ t3, int N) {
  int n = blockIdx.x * blockDim.x + threadIdx.x;
  if (n >= N) return;
  const float* hr = h + (size_t)n * 16;
  float a0 = 0.f, a1 = 0.f, a2 = 0.f;
#pragma unroll
  for (int k = 0; k < 16; ++k) {
    float hv = hr[k];
    a0 += hv * W3[k * 3 + 0];
    a1 += hv * W3[k * 3 + 1];
    a2 += hv * W3[k * 3 + 2];
  }
  t3[(size_t)n * 3 + 0] = a0;
  t3[(size_t)n * 3 + 1] = a1;
  t3[(size_t)n * 3 + 2] = a2;
}

// ---- self-loop init: out[n,f] = t[n,f] * dinv[n%V]^2 ----
__global__ void self_init_k(const float* __restrict__ t, float* __restrict__ out,
                            const float* __restrict__ dinv, long long total, int F, int Vn) {
  long long i = blockIdx.x * (long long)blockDim.x + threadIdx.x;
  if (i >= total) return;
  int node = (int)(i / F);
  float dv = dinv[node % Vn];
  out[i] = t[i] * dv * dv;
}

// ---- edge scatter, 16 features, one thread per (batch, edge) ----
__global__ void scatter16_k(const float* __restrict__ t, float* __restrict__ out,
                            const int* __restrict__ edges, const float* __restrict__ dinv,
                            int E, int Vn, int B) {
  long long tid = blockIdx.x * (long long)blockDim.x + threadIdx.x;
  if (tid >= (long long)B * E) return;
  int e = (int)(tid % E);
  int b = (int)(tid / E);
  int s = edges[2 * e], d = edges[2 * e + 1];
  float nrm = dinv[s] * dinv[d];
  long long sg = (long long)b * Vn + s;
  long long dg = (long long)b * Vn + d;
  const float4* ts = (const float4*)(t + sg * 16);
  float* od = out + dg * 16;
#pragma unroll
  for (int q = 0; q < 4; ++q) {
    float4 v = ts[q];
    atomAddF(od + 4 * q + 0, v.x * nrm);
    atomAddF(od + 4 * q + 1, v.y * nrm);
    atomAddF(od + 4 * q + 2, v.z * nrm);
    atomAddF(od + 4 * q + 3, v.w * nrm);
  }
}

// ---- edge scatter, 3 features ----
__global__ void scatter3_k(const float* __restrict__ t, float* __restrict__ out,
                           const int* __restrict__ edges, const float* __restrict__ dinv,
                           int E, int Vn, int B) {
  long long tid = blockIdx.x * (long long)blockDim.x + threadIdx.x;
  if (tid >= (long long)B * E) return;
  int e = (int)(tid % E);
  int b = (int)(tid / E);
  int s = edges[2 * e], d = edges[2 * e + 1];
  float nrm = dinv[s] * dinv[d];
  long long sg = (long long)b * Vn + s;
  long long dg = (long long)b * Vn + d;
#pragma unroll
  for (int c = 0; c < 3; ++c) atomAddF(out + dg * 3 + c, t[sg * 3 + c] * nrm);
}

// ---- bias + leaky relu ----
__global__ void finalize_k(float* __restrict__ out, const float* __restrict__ bias,
                           long long total, int F) {
  long long i = blockIdx.x * (long long)blockDim.x + threadIdx.x;
  if (i >= total) return;
  float v = out[i] + bias[(int)(i % F)];
  out[i] = v > 0.f ? v : NEG_SLOPE * v;
}

// ---- v = x + offs + b3, scattered into d_out layout (B, V+E, 3) ----
__global__ void verts_out_k(const float* __restrict__ x, const float* __restrict__ o3,
                            const float* __restrict__ b3, float* __restrict__ vout,
                            int N, int Vn, int E) {
  int n = blockIdx.x * blockDim.x + threadIdx.x;
  if (n >= N) return;
  int b = n / Vn, i = n % Vn;
  float* dst = vout + ((size_t)b * (Vn + E) + i) * 3;
#pragma unroll
  for (int c = 0; c < 3; ++c)
    dst[c] = x[(size_t)n * 3 + c] + o3[(size_t)n * 3 + c] + b3[c];
}

// ---- edge midpoints from v region of d_out ----
__global__ void mid_k(const int* __restrict__ edges, float* __restrict__ vout,
                      int E, int Vn, int B) {
  long long tid = blockIdx.x * (long long)blockDim.x + threadIdx.x;
  if (tid >= (long long)B * E) return;
  int e = (int)(tid % E);
  int b = (int)(tid / E);
  int s = edges[2 * e], d = edges[2 * e + 1];
  const float* vb = vout + (size_t)b * (Vn + E) * 3;
  float* m = vout + ((size_t)b * (Vn + E) + Vn + e) * 3;
#pragma unroll
  for (int c = 0; c < 3; ++c) m[c] = 0.5f * (vb[(size_t)s * 3 + c] + vb[(size_t)d * 3 + c]);
}

// ---- broadcast faces (int32 bits) ----
__global__ void faces_k(const int* __restrict__ faces, int* __restrict__ fo,
                        int FS3, long long total) {
  long long i = blockIdx.x * (long long)blockDim.x + threadIdx.x;
  if (i >= total) return;
  fo[i] = faces[(int)(i % FS3)];
}

static inline int cdiv(long long a, long long b) { return (int)((a + b - 1) / b); }

extern "C" void kernel_launch(void* const* d_in, const int* in_sizes, int n_in,
                              void* d_out, int out_size, void* d_ws, size_t ws_size,
                              hipStream_t stream) {
  const float* verts = (const float*)d_in[0];
  const int*   edges = (const int*)d_in[1];
  const int*   faces = (const int*)d_in[2];
  const float* W1 = (const float*)d_in[3];
  const float* b1 = (const float*)d_in[4];
  const float* W2 = (const float*)d_in[5];
  const float* b2 = (const float*)d_in[6];
  const float* W3 = (const float*)d_in[7];
  const float* b3 = (const float*)d_in[8];

  const int B   = 16;                       // fixed by setup_inputs
  const int Vn  = in_sizes[0] / (B * 3);    // 100000
  const int E   = in_sizes[1] / 2;          // 300000
  const int FS3 = in_sizes[2];              // Fsub*3
  const int N   = B * Vn;                   // 1.6M (divisible by 16)

  // workspace: T (N*16 f32) | H (N*16 f32) | dinv (V f32) | deg (V i32)
  float* T  = (float*)d_ws;
  float* Hb = T + (size_t)N * 16;
  float* DI = Hb + (size_t)N * 16;
  int*   DEG = (int*)(DI + Vn);
  float* T3 = T;                    // conv3 reuses dead T region: N*3
  float* O3 = T + (size_t)N * 3;    // N*3, disjoint from T3

  float* vout = (float*)d_out;                            // (B, V+E, 3)
  int*   fout = (int*)(vout + (size_t)B * (Vn + E) * 3);  // (B, Fsub, 3)

  const int TB = 256;
  const long long NE = (long long)B * E;
  const long long N16 = (long long)N * 16;
  const int mmBlocks = cdiv(N / 16, 4);     // 4 waves/block, one 16-row tile each

  // normalization
  hipMemsetAsync(DEG, 0, sizeof(int) * Vn, stream);
  count_deg_k<<<cdiv(E, TB), TB, 0, stream>>>(edges, DEG, E);
  dinv_k<<<cdiv(Vn, TB), TB, 0, stream>>>(DEG, DI, Vn);

  // conv1 + leaky
  conv1_wmma_k<<<mmBlocks, 128, 0, stream>>>(verts, W1, T, N);
  self_init_k<<<cdiv(N16, TB), TB, 0, stream>>>(T, Hb, DI, N16, 16, Vn);
  scatter16_k<<<cdiv(NE, TB), TB, 0, stream>>>(T, Hb, edges, DI, E, Vn, B);
  finalize_k<<<cdiv(N16, TB), TB, 0, stream>>>(Hb, b1, N16, 16);

  // conv2 + leaky
  conv2_wmma_k<<<mmBlocks, 128, 0, stream>>>(Hb, W2, T, N);
  self_init_k<<<cdiv(N16, TB), TB, 0, stream>>>(T, Hb, DI, N16, 16, Vn);
  scatter16_k<<<cdiv(NE, TB), TB, 0, stream>>>(T, Hb, edges, DI, E, Vn, B);
  finalize_k<<<cdiv(N16, TB), TB, 0, stream>>>(Hb, b2, N16, 16);

  // conv3 (bias fused into verts_out)
  conv3_k<<<cdiv(N, TB), TB, 0, stream>>>(Hb, W3, T3, N);
  self_init_k<<<cdiv((long long)N * 3, TB), TB, 0, stream>>>(T3, O3, DI, (long long)N * 3, 3, Vn);
  scatter3_k<<<cdiv(NE, TB), TB, 0, stream>>>(T3, O3, edges, DI, E, Vn, B);

  // subdivision epilogue
  verts_out_k<<<cdiv(N, TB), TB, 0, stream>>>(verts, O3, b3, vout, N, Vn, E);
  mid_k<<<cdiv(NE, TB), TB, 0, stream>>>(edges, vout, E, Vn, B);
  faces_k<<<cdiv((long long)B * FS3, TB), TB, 0, stream>>>(faces, fout, FS3, (long long)B * FS3);
}